// QuantizedAttentionMapOpenSORA_86028194939399
// MI455X (gfx1250) — compile-verified
//
#include <hip/hip_runtime.h>
#include <stdint.h>

typedef __attribute__((ext_vector_type(4))) unsigned int uint32x4;
typedef __attribute__((ext_vector_type(8))) int          int32x8;
typedef __attribute__((ext_vector_type(4))) int          int32x4;

#define N_TOK  5520
#define N_TEXT 120
#define N_IMG  5400
#define QMAXF  127.0f
#define HEAD_STRIDE ((size_t)N_TOK * (size_t)N_TOK)

__device__ __forceinline__ float fake_quant(float v, float delta) {
    float d = delta > 0.0f ? delta : 1.0f;
    float scale = d / QMAXF;
    float q = rintf(v / scale);               // round-half-to-even, matches jnp.round
    q = fminf(fmaxf(q, -QMAXF), QMAXF);
    return q * scale;
}

// ---------------- workspace init (head-0 block maxima) ----------------
__global__ void init_ws_kernel(float* ws, int n) {
    int i = blockIdx.x * blockDim.x + threadIdx.x;
    if (i < n) ws[i] = 0.0f;
}

// ---------------- head 0: pass 1, per-block max (bw=1350, 4x4 blocks) ----------------
__global__ void head0_block_max(const float* __restrict__ x, float* __restrict__ ws) {
    __shared__ float red[256];
    const int BW = 1350;
    const int blk = blockIdx.y;                 // 0..15
    const int by = blk >> 2, bx = blk & 3;
    const float* base = x + (size_t)(N_TEXT + by * BW) * N_TOK + (N_TEXT + bx * BW);

    float m = 0.0f;
    for (int r = blockIdx.x; r < BW; r += gridDim.x) {
        const float* row = base + (size_t)r * N_TOK;
        for (int c = threadIdx.x; c < BW; c += blockDim.x)
            m = fmaxf(m, row[c]);
    }
    red[threadIdx.x] = m;
    __syncthreads();
    for (int s = 128; s > 0; s >>= 1) {
        if (threadIdx.x < s) red[threadIdx.x] = fmaxf(red[threadIdx.x], red[threadIdx.x + s]);
        __syncthreads();
    }
    // nonnegative floats order-preserve as uint bit patterns
    if (threadIdx.x == 0)
        atomicMax((unsigned int*)(ws + blk), __float_as_uint(red[0]));
}

// ---------------- head 0: pass 2, quantize (float4, per-lane block lookup) ----------------
__global__ void head0_quant(const float* __restrict__ x, float* __restrict__ out,
                            const float* __restrict__ ws) {
    const int BW = 1350;
    const int r  = blockIdx.x;                  // 0..5399
    const int by = r / BW;
    const float* src = x  + (size_t)(N_TEXT + r) * N_TOK + N_TEXT;
    float*       dst = out + (size_t)(N_TEXT + r) * N_TOK + N_TEXT;

    for (int c4 = threadIdx.x; c4 < N_IMG / 4; c4 += blockDim.x) {
        int c = c4 * 4;
        float4 v = *(const float4*)(src + c);
        float q0 = fake_quant(v.x, ws[by * 4 + (c + 0) / BW]);
        float q1 = fake_quant(v.y, ws[by * 4 + (c + 1) / BW]);
        float q2 = fake_quant(v.z, ws[by * 4 + (c + 2) / BW]);
        float q3 = fake_quant(v.w, ws[by * 4 + (c + 3) / BW]);
        *(float4*)(dst + c) = make_float4(q0, q1, q2, q3);
    }
}

// ---------------- pass-through copies (text rows / text cols, all heads) ----------------
__global__ void copy_text_rows(const float* __restrict__ x, float* __restrict__ out) {
    const size_t per_head4 = (size_t)N_TEXT * N_TOK / 4;       // contiguous float4's
    const size_t total     = per_head4 * 4;
    for (size_t i = (size_t)blockIdx.x * blockDim.x + threadIdx.x; i < total;
         i += (size_t)gridDim.x * blockDim.x) {
        size_t h = i / per_head4, j = i % per_head4;
        size_t off = h * (HEAD_STRIDE / 4) + j;
        ((float4*)out)[off] = ((const float4*)x)[off];
    }
}

__global__ void copy_text_cols(const float* __restrict__ x, float* __restrict__ out) {
    const int W4 = N_TEXT / 4;                                  // 30 float4 per row
    const size_t per_head = (size_t)N_IMG * W4;
    const size_t total    = per_head * 4;
    for (size_t i = (size_t)blockIdx.x * blockDim.x + threadIdx.x; i < total;
         i += (size_t)gridDim.x * blockDim.x) {
        size_t h = i / per_head, rem = i % per_head;
        size_t r = rem / W4, c4 = rem % W4;
        size_t off = h * HEAD_STRIDE + (N_TEXT + r) * (size_t)N_TOK + c4 * 4;
        *(float4*)(out + off) = *(const float4*)(x + off);
    }
}

// ---------------- heads 1-3: TDM tile in LDS -> max -> quantize -> TDM tile out ----------
// One workgroup (8 wave32's) per quant block; block fits in the 320KB WGP LDS.
__global__ __launch_bounds__(256)
void tdm_block_quant(const float* __restrict__ src_head, float* __restrict__ dst_head,
                     int bw, int nb) {
    extern __shared__ float tile[];
    __shared__ float red[256];

    const int tid = threadIdx.x;
    const int blk = blockIdx.x;
    const int by = blk / nb, bx = blk % nb;
    const int total = bw * bw;

    const size_t elem_off = (size_t)(N_TEXT + by * bw) * N_TOK + (size_t)(N_TEXT + bx * bw);
    const uint64_t gsrc = (uint64_t)(uintptr_t)(src_head + elem_off);
    const uint64_t gdst = (uint64_t)(uintptr_t)(dst_head + elem_off);
    const uint32_t lds_addr = (uint32_t)(uintptr_t)tile;        // LDS byte offset (addrspace 3)

    // ---- D# group 1 (256b): mask=0, data_size=4B(2), 2D dims, 48-bit row stride ----
    int32x8 g1;
    g1[0] = 0x00020000;                                         // data_size = 2 -> 4 bytes
    g1[1] = (int)(((uint32_t)bw & 0xFFFFu) << 16);              // tensor_dim0[15:0] @ bits 63:48
    g1[2] = (int)((((uint32_t)bw >> 16) & 0xFFFFu) |            // tensor_dim0[31:16]
                  (((uint32_t)bw & 0xFFFFu) << 16));            // tensor_dim1[15:0]
    g1[3] = (int)((((uint32_t)bw >> 16) & 0xFFFFu) |            // tensor_dim1[31:16]
                  (((uint32_t)bw & 0xFFFFu) << 16));            // tile_dim0
    g1[4] = (int)((uint32_t)bw & 0xFFFFu);                      // tile_dim1 (tile_dim2 = 0)
    g1[5] = N_TOK;                                              // tensor_dim0_stride[31:0] = 5520
    g1[6] = 0;                                                  // stride hi / dim1_stride lo
    g1[7] = 0;
    const int32x4 gz4 = {0, 0, 0, 0};                           // groups 2/3 unused (2D tensor)
    const int32x8 gz8 = {0, 0, 0, 0, 0, 0, 0, 0};               // extra group (6-arg form)

    if (tid == 0) {
        uint32x4 g0;
        g0[0] = 1u;                                             // count=1, user descriptor
        g0[1] = lds_addr;
        g0[2] = (uint32_t)(gsrc & 0xFFFFFFFFu);
        g0[3] = (uint32_t)((gsrc >> 32) & 0x1FFFFFFu) | (2u << 30);   // type=2 (image)
        __builtin_amdgcn_tensor_load_to_lds(g0, g1, gz4, gz4, gz8, 0);
        __builtin_amdgcn_s_wait_tensorcnt(0);
    }
    __syncthreads();

    // block max out of LDS
    float m = 0.0f;
    for (int i = tid; i < total; i += 256) m = fmaxf(m, tile[i]);
    red[tid] = m;
    __syncthreads();
    for (int s = 128; s > 0; s >>= 1) {
        if (tid < s) red[tid] = fmaxf(red[tid], red[tid + s]);
        __syncthreads();
    }
    const float delta = red[0];
    __syncthreads();

    // fake-quantize in place in LDS
    for (int i = tid; i < total; i += 256) tile[i] = fake_quant(tile[i], delta);
    __syncthreads();

    if (tid == 0) {
        uint32x4 g0;
        g0[0] = 1u;
        g0[1] = lds_addr;
        g0[2] = (uint32_t)(gdst & 0xFFFFFFFFu);
        g0[3] = (uint32_t)((gdst >> 32) & 0x1FFFFFFu) | (2u << 30);
        __builtin_amdgcn_tensor_store_from_lds(g0, g1, gz4, gz4, gz8, 0);
        __builtin_amdgcn_s_wait_tensorcnt(0);
    }
}

extern "C" void kernel_launch(void* const* d_in, const int* in_sizes, int n_in,
                              void* d_out, int out_size, void* d_ws, size_t ws_size,
                              hipStream_t stream) {
    const float* x  = (const float*)d_in[0];     // [1, 4, 5520, 5520] f32
    float* out = (float*)d_out;
    float* ws  = (float*)d_ws;                   // 16 floats: head-0 block maxima

    // head 0 (bw=1350, blocks too big for LDS): two-pass
    init_ws_kernel<<<1, 32, 0, stream>>>(ws, 16);
    head0_block_max<<<dim3(128, 16), 256, 0, stream>>>(x, ws);
    head0_quant<<<N_IMG, 256, 0, stream>>>(x, out, ws);

    // pass-through text regions, all heads
    copy_text_rows<<<1024, 256, 0, stream>>>(x, out);
    copy_text_cols<<<1024, 256, 0, stream>>>(x, out);

    // heads 1-3: TDM-staged single-pass per block
    const int chunks[3] = {27, 54, 108};
    for (int h = 1; h < 4; ++h) {
        const int nb = chunks[h - 1];
        const int bw = N_IMG / nb;               // 200 / 100 / 50
        const size_t shmem = (size_t)bw * bw * sizeof(float);   // 160KB / 40KB / 10KB
        tdm_block_quant<<<nb * nb, 256, shmem, stream>>>(
            x + (size_t)h * HEAD_STRIDE, out + (size_t)h * HEAD_STRIDE, bw, nb);
    }
}